// ReconDetectionConfidenceMap_74079595922147
// MI455X (gfx1250) — compile-verified
//
#include <hip/hip_runtime.h>
#include <cstdint>

// ReconDetectionConfidenceMap for MI455X (gfx1250, wave32).
//
// Shapes: B=16, K=64, H=160, W=160.  out[b,k,i,j] = den*exp(num1*num2(b,k,i,j))
// normalized over k.  DetectionMap is shape-only (never read).
//
// Roofline: writes 16*64*160*160*4 B = 104.9 MB, reads ~8 KB -> 4.5 us floor
// at 23.3 TB/s.  Strategy: one thread per pixel (b,i,j); fully-unrolled K=64
// loop keeps all 64 scores in VGPRs, so exp is evaluated exactly once per
// output element and scoreMap never touches HBM before normalization.
// Keypoints (512 B per batch) are staged to LDS with the CDNA5 async
// global->LDS path (one global_load_async_to_lds_b128 from wave 0, waited
// with s_wait_asynccnt), then broadcast-read from LDS by all 8 waves.

#define BATCH 16
#define KNUM  64
#define HNUM  160
#define WNUM  160
#define HWSZ  (HNUM * WNUM)     // 25600
#define BDIM  256               // 8 wave32 waves; 25600 % 256 == 0 -> block never crosses a batch
#define EPSV  1e-5f

__global__ __launch_bounds__(BDIM) void recon_conf_map_kernel(
    const float* __restrict__ keypoints,   // [B, K, 2] (x, y) interleaved
    const float* __restrict__ p_sx,
    const float* __restrict__ p_sy,
    const float* __restrict__ p_rho,
    float* __restrict__ out)               // [B, K, H, W]
{
    __shared__ __align__(16) float lds_kp[2 * KNUM];   // 512 B: (x,y) pairs for this batch

    const int p = blockIdx.x * BDIM + threadIdx.x;     // global pixel id
    const int b = blockIdx.x / (HWSZ / BDIM);          // 100 blocks per batch (uniform per block)

    // ---- CDNA5 async stage: 32 lanes x 16 B = 512 B of keypoints -> LDS ----
    if (threadIdx.x < 32) {
        const uint32_t lane = threadIdx.x;
        // Flat LDS addresses truncate to addr[31:0] == LDS byte offset (ISA 10.2).
        uint32_t lds_addr = (uint32_t)(uintptr_t)(&lds_kp[0]) + lane * 16u;
        uint64_t gaddr    = (uint64_t)(uintptr_t)(keypoints + (size_t)b * (2 * KNUM))
                          + (uint64_t)lane * 16u;
        asm volatile("global_load_async_to_lds_b128 %0, %1, off"
                     :: "v"(lds_addr), "v"(gaddr) : "memory");
        asm volatile("s_wait_asynccnt 0" ::: "memory");
    }
    __syncthreads();

    // ---- per-thread constants (scalar inputs are 1-element device arrays) ----
    const float sx  = *p_sx;
    const float sy  = *p_sy;
    const float rho = *p_rho;

    const float omr2 = 1.0f - rho * rho;
    const float num1 = -0.5f / omr2;
    const float den  = 0.15915494309189535f / (sx * sy * __builtin_sqrtf(omr2)); // 1/(2*pi*sx*sy*sqrt(1-rho^2))
    const float isy2 = 1.0f / (sy * sy);
    const float isx2 = 1.0f / (sx * sx);
    const float cr   = 2.0f * rho / (sx * sy);

    const int j = p % WNUM;
    const int i = (p / WNUM) % HNUM;
    const float fi = (float)i;
    const float fj = (float)j;

    // ---- single pass: 64 scores live in VGPRs, denominator accumulated ----
    float scores[KNUM];
    float sum = 0.0f;
#pragma unroll
    for (int k = 0; k < KNUM; ++k) {
        const float kx = lds_kp[2 * k + 0];   // uniform address -> LDS broadcast
        const float ky = lds_kp[2 * k + 1];
        const float dy = fi - ky;
        const float dx = fj - kx;
        float n2 = dy * dy * isy2;
        n2 = __builtin_fmaf(dx * dx, isx2, n2);
        n2 = __builtin_fmaf(-cr * dy, dx, n2);
        const float s = den * __expf(num1 * n2);  // v_exp_f32 fast path
        scores[k] = s;
        sum += s;
    }

    const float r = 1.0f / (sum + EPSV);          // accurate division (rcp + Newton)

    // ---- 64 coalesced store streams: lanes contiguous in j, stride H*W in k ----
    float* o = out + (size_t)b * ((size_t)KNUM * HWSZ) + (size_t)i * WNUM + (size_t)j;
#pragma unroll
    for (int k = 0; k < KNUM; ++k) {
        o[(size_t)k * HWSZ] = scores[k] * r;
    }
}

extern "C" void kernel_launch(void* const* d_in, const int* in_sizes, int n_in,
                              void* d_out, int out_size, void* d_ws, size_t ws_size,
                              hipStream_t stream) {
    (void)in_sizes; (void)n_in; (void)out_size; (void)d_ws; (void)ws_size;
    const float* keypoints = (const float*)d_in[0];  // [B,K,2]
    const float* sx        = (const float*)d_in[1];  // scalar
    const float* sy        = (const float*)d_in[2];  // scalar
    const float* rho       = (const float*)d_in[3];  // scalar
    // d_in[4] = DetectionMap: shape-only in the reference, never read.
    float* out = (float*)d_out;

    const int total_pixels = BATCH * HNUM * WNUM;    // 409,600
    dim3 grid(total_pixels / BDIM);                  // 1600 blocks
    recon_conf_map_kernel<<<grid, BDIM, 0, stream>>>(keypoints, sx, sy, rho, out);
}